// Sparse2to4Linear_78486232367564
// MI455X (gfx1250) — compile-verified
//
#include <hip/hip_runtime.h>

#if defined(__AMDGCN__) && __has_builtin(__builtin_amdgcn_tensor_load_to_lds) && \
    __has_builtin(__builtin_amdgcn_s_wait_tensorcnt)
#define S24_TDM 1
#else
#define S24_TDM 0
#endif

typedef __attribute__((ext_vector_type(4)))  float          f32x4;
typedef __attribute__((ext_vector_type(8)))  float          v8f;
typedef __attribute__((ext_vector_type(16))) __bf16         v16bf;
typedef __attribute__((ext_vector_type(32))) __bf16         v32bf;
typedef __attribute__((ext_vector_type(4)))  unsigned short u16x4;
typedef __attribute__((ext_vector_type(8)))  unsigned short u16x8;
typedef __attribute__((ext_vector_type(16))) unsigned short u16x16;
typedef __attribute__((ext_vector_type(32))) unsigned short u16x32;

static __device__ __forceinline__ unsigned short f32_to_bf16(float f) {
    unsigned int u = __builtin_bit_cast(unsigned int, f);
    u += 0x7fffu + ((u >> 16) & 1u);   // round-to-nearest-even
    return (unsigned short)(u >> 16);
}

// 2:4 pick: top-2 magnitudes (lowest-index tie-break), soft-shrunk values,
// 4-bit code {idx1[1:0], idx0[1:0]} with idx0 < idx1 (ISA requirement).
static __device__ __forceinline__ void pick24(f32x4 w, float& v0, float& v1,
                                              unsigned& code) {
    float a0 = fabsf(w.x), a1 = fabsf(w.y), a2 = fabsf(w.z), a3 = fabsf(w.w);
    float lo01 = fminf(a0, a1), hi01 = fmaxf(a0, a1);
    float lo23 = fminf(a2, a3), hi23 = fmaxf(a2, a3);
    float t = fminf(fmaxf(lo01, lo23), fminf(hi01, hi23)); // 2nd smallest of 4
    int j0 = 0; float mv = a0;
    if (a1 > mv) { mv = a1; j0 = 1; }
    if (a2 > mv) { mv = a2; j0 = 2; }
    if (a3 > mv) { mv = a3; j0 = 3; }
    int j1; float m2;
    if (j0 == 0) { j1 = 1; m2 = a1; } else { j1 = 0; m2 = a0; }
    if (j0 != 1 && a1 > m2) { j1 = 1; m2 = a1; }
    if (j0 != 2 && a2 > m2) { j1 = 2; m2 = a2; }
    if (j0 != 3 && a3 > m2) { j1 = 3; m2 = a3; }
    int i0 = (j0 < j1) ? j0 : j1;
    int i1 = (j0 < j1) ? j1 : j0;
    float s0 = (i0 == 0) ? w.x : (i0 == 1) ? w.y : (i0 == 2) ? w.z : w.w;
    float s1 = (i1 == 1) ? w.y : (i1 == 2) ? w.z : w.w;   // i1 >= 1 always
    float b0 = fabsf(s0), b1 = fabsf(s1);
    v0 = (b0 > t) ? copysignf(b0 - t, s0) : 0.0f;
    v1 = (b1 > t) ? copysignf(b1 - t, s1) : 0.0f;
    code = (unsigned)((i1 << 2) | i0);
}

#if S24_TDM
typedef __attribute__((ext_vector_type(4))) unsigned int u32x4;
typedef __attribute__((ext_vector_type(8))) int          i32x8;
typedef __attribute__((ext_vector_type(4))) int          i32x4;

// 2D TDM load: tile_d1 rows of tile_d0 elements, row pitch = stride elems,
// with LDS padding pad_amount after every pad_interval (codes per D# spec).
static __device__ __forceinline__ void tdm_load_2d(
    const void* gptr, unsigned lds_addr, unsigned tile_d0, unsigned tile_d1,
    unsigned stride_elems, unsigned data_size_code,
    unsigned pad_interval_code, unsigned pad_amount_code) {
    unsigned long long ga = (unsigned long long)gptr;
    u32x4 g0;
    g0.x = 1u;                                                // count=1 (valid)
    g0.y = lds_addr;                                          // LDS byte addr
    g0.z = (unsigned)ga;                                      // gaddr[31:0]
    g0.w = (unsigned)((ga >> 32) & 0x01ffffffu) | (2u << 30); // [56:32]|type=2
    i32x8 g1;
    g1[0] = (int)((data_size_code << 16) | (1u << 20) |       // pad_enable
                  (pad_interval_code << 22) | (pad_amount_code << 25));
    g1[1] = (int)((tile_d0 & 0xffffu) << 16);                 // tensor_dim0 lo
    g1[2] = (int)(((tile_d0 >> 16) & 0xffffu) |               // tensor_dim0 hi
                  ((tile_d1 & 0xffffu) << 16));               // tensor_dim1 lo
    g1[3] = (int)(((tile_d1 >> 16) & 0xffffu) |               // tensor_dim1 hi
                  ((tile_d0 & 0xffffu) << 16));               // tile_dim0
    g1[4] = (int)(tile_d1 & 0xffffu);                         // tile_dim1
    g1[5] = (int)stride_elems;                                // dim0_stride lo
    g1[6] = 0;
    g1[7] = 0;
    i32x4 z4 = {};
#if __clang_major__ >= 23
    i32x8 z8 = {};
    __builtin_amdgcn_tensor_load_to_lds(g0, g1, z4, z4, z8, 0);
#else
    __builtin_amdgcn_tensor_load_to_lds(g0, g1, z4, z4, 0);
#endif
}
#endif

// ---- pack W once: 2:4 bf16 values + index words + fused scale partials ----
// One thread = one (row, 32-K segment) = 8 groups. Block = 2 rows.
// Index words stored K-major: pidx[(k/64)*2*M + row*2 + half].

__global__ __launch_bounds__(256) void s24_pack_kernel(
    const float* __restrict__ w, unsigned short* __restrict__ pw,
    unsigned* __restrict__ pidx, float* __restrict__ pd, float* __restrict__ ps,
    int K, int M) {
    __shared__ float sd[256];
    __shared__ float ss[256];
    const int tid = threadIdx.x;
    const int row = blockIdx.x * 2 + (tid >> 7);
    const int seg = tid & 127;                // 32-K segment within row
    const float* src = w + (size_t)row * K + seg * 32;

    u16x8 p0 = {}, p1 = {};
    unsigned codew = 0;
    float d2 = 0.0f, s2 = 0.0f;
#pragma unroll
    for (int g = 0; g < 8; ++g) {
        f32x4 v = *(const f32x4*)(src + g * 4);
        float v0f, v1f; unsigned code;
        pick24(v, v0f, v1f, code);
        d2 += v.x * v.x + v.y * v.y + v.z * v.z + v.w * v.w;
        s2 += v0f * v0f + v1f * v1f;          // others are exactly 0
        unsigned short h0 = f32_to_bf16(v0f);
        unsigned short h1 = f32_to_bf16(v1f);
        if (g < 4) { p0[2 * g] = h0;       p0[2 * g + 1] = h1; }
        else       { p1[2 * (g - 4)] = h0; p1[2 * (g - 4) + 1] = h1; }
        codew |= code << (4 * g);
    }
    unsigned short* dst = pw + (size_t)row * (K >> 1) + seg * 16;
    *(u16x8*)dst       = p0;
    *(u16x8*)(dst + 8) = p1;
    pidx[(size_t)(seg >> 1) * (2 * (size_t)M) + row * 2 + (seg & 1)] = codew;

    sd[tid] = d2; ss[tid] = s2;
    __syncthreads();
    for (int off = 128; off > 0; off >>= 1) {
        if (tid < off) { sd[tid] += sd[tid + off]; ss[tid] += ss[tid + off]; }
        __syncthreads();
    }
    if (tid == 0) { pd[blockIdx.x] = sd[0]; ps[blockIdx.x] = ss[0]; }
}

__global__ __launch_bounds__(256) void s24_scale_finalize(
    const float* __restrict__ pd, const float* __restrict__ ps, int n,
    float* __restrict__ scale_out) {
    __shared__ float sd[256];
    __shared__ float ss[256];
    const int tid = threadIdx.x;
    float d = 0.0f, s = 0.0f;
    for (int i = tid; i < n; i += 256) { d += pd[i]; s += ps[i]; }
    sd[tid] = d; ss[tid] = s;
    __syncthreads();
    for (int off = 128; off > 0; off >>= 1) {
        if (tid < off) { sd[tid] += sd[tid + off]; ss[tid] += ss[tid + off]; }
        __syncthreads();
    }
    if (tid == 0) {
        float sc = sqrtf(sd[0] / fmaxf(ss[0], 1e-12f));
        scale_out[0] = fminf(fmaxf(sc, 0.1f), 10.0f);
    }
}

// ---- optional: x -> bf16 precompute (one float4 per thread) ----

__global__ __launch_bounds__(256) void s24_xcvt_kernel(
    const float* __restrict__ src, unsigned short* __restrict__ dst) {
    size_t i = ((size_t)blockIdx.x * 256 + threadIdx.x) * 4;
    f32x4 v = *(const f32x4*)(src + i);
    u16x4 h;
    h.x = f32_to_bf16(v.x); h.y = f32_to_bf16(v.y);
    h.z = f32_to_bf16(v.z); h.w = f32_to_bf16(v.w);
    *(u16x4*)(dst + i) = h;
}

// -------- 2:4 sparse GEMM (SWMMAC): y^T = ws . x^T, + bias ----------------
// Double-buffered; XB16 path stages tiles with the Tensor Data Mover.

#define TF 128   // out-features per block  (A/M dimension)
#define TT 128   // tokens per block        (B/N dimension)
#define KS 64    // K per step (one swmmac)
#define APS 40   // packed-A row stride (halves): 32 data + 8 pad (TDM pad 3/3)
#define BPS 72   // B row stride (halves): 64 data + 8 pad (TDM pad 4/3)

template <bool XB16>
__global__ __launch_bounds__(256) void s24_swmmac_gemm(
    const float* __restrict__ xf, const unsigned short* __restrict__ xh,
    const unsigned short* __restrict__ pw, const unsigned* __restrict__ pidx,
    const float* __restrict__ bias, const float* __restrict__ scale_p,
    float* __restrict__ out, int N, int K) {
    __shared__ __align__(16) unsigned short As[2][TF * APS];
    __shared__ unsigned AI[2][TF * 2];
    __shared__ __align__(16) unsigned short Bs[2][TT * BPS];

    const int tid   = threadIdx.x;
    const int lane  = tid & 31;
    const int wave  = tid >> 5;
    const int wm    = wave >> 2;
    const int wn    = wave & 3;
    const int F0    = blockIdx.x * TF;
    const int T0    = blockIdx.y * TT;
    const int lrow  = lane & 15;
    const int lhalf = lane >> 4;
    const float scale = scale_p[0];
    const int nk = K / KS;

    const float* xfb          = xf + (size_t)T0 * K;
    const unsigned short* xhb = xh + (size_t)T0 * K;

    // index words, K-major: one coalesced u32 per thread per step
    const unsigned* pai = pidx + (size_t)F0 * 2 + tid;
    unsigned pi;
    auto loadI  = [&](int k0) { pi = pai[(size_t)(k0 >> 6) * (2 * (size_t)N)]; };
    auto storeI = [&](int b)  { AI[b][tid] = pi; };

    // manual staging (fallback / fp32-x path)
    const int arow = tid >> 2, ach = tid & 3;
    const unsigned short* paw = pw + (size_t)(F0 + arow) * (K >> 1) + ach * 8;
    const size_t pawStep = (size_t)64 * (K >> 1);
    u16x8 pa0, pa1;
    u16x8 pbh[4];
    f32x4 pbf[8];

    auto loadA = [&](int k0) {
        pa0 = *(const u16x8*)(paw + (k0 >> 1));
        pa1 = *(const u16x8*)(paw + pawStep + (k0 >> 1));
    };
    auto storeA = [&](int b) {
        *(u16x8*)(&As[b][arow * APS + ach * 8])        = pa0;
        *(u16x8*)(&As[b][(arow + 64) * APS + ach * 8]) = pa1;
    };
    auto loadB = [&](int k0) {
        if (XB16) {
#pragma unroll
            for (int it = 0; it < 4; ++it) {
                int c = tid + it * 256;
                int row = c >> 3, ch = c & 7;
                pbh[it] = *(const u16x8*)(xhb + (size_t)row * K + k0 + ch * 8);
            }
        } else {
#pragma unroll
            for (int it = 0; it < 8; ++it) {
                int c = tid + it * 256;
                int row = c >> 4, c4 = c & 15;
                pbf[it] = *(const f32x4*)(xfb + (size_t)row * K + k0 + c4 * 4);
            }
        }
    };
    auto storeB = [&](int b) {
        if (XB16) {
#pragma unroll
            for (int it = 0; it < 4; ++it) {
                int c = tid + it * 256;
                int row = c >> 3, ch = c & 7;
                *(u16x8*)(&Bs[b][row * BPS + ch * 8]) = pbh[it];
            }
        } else {
#pragma unroll
            for (int it = 0; it < 8; ++it) {
                int c = tid + it * 256;
                int row = c >> 4, c4 = c & 15;
                u16x4 h;
                h.x = f32_to_bf16(pbf[it].x); h.y = f32_to_bf16(pbf[it].y);
                h.z = f32_to_bf16(pbf[it].z); h.w = f32_to_bf16(pbf[it].w);
                *(u16x4*)(&Bs[b][row * BPS + c4 * 4]) = h;
            }
        }
    };

    v8f acc[4][2];
#pragma unroll
    for (int t = 0; t < 4; ++t)
#pragma unroll
        for (int u = 0; u < 2; ++u) acc[t][u] = {};

    auto compute = [&](int b) {
        v32bf bfrag[2];
#pragma unroll
        for (int u = 0; u < 2; ++u) {
            const unsigned short* bp =
                &Bs[b][(wn * 32 + u * 16 + lrow) * BPS + lhalf * 16];
            u16x8 c0 = *(const u16x8*)bp;
            u16x8 c1 = *(const u16x8*)(bp + 8);
            u16x8 c2 = *(const u16x8*)(bp + 32);
            u16x8 c3 = *(const u16x8*)(bp + 40);
            u16x16 blo = __builtin_shufflevector(c0, c1,
                0, 1, 2, 3, 4, 5, 6, 7, 8, 9, 10, 11, 12, 13, 14, 15);
            u16x16 bhi = __builtin_shufflevector(c2, c3,
                0, 1, 2, 3, 4, 5, 6, 7, 8, 9, 10, 11, 12, 13, 14, 15);
            u16x32 bb = __builtin_shufflevector(blo, bhi,
                0, 1, 2, 3, 4, 5, 6, 7, 8, 9, 10, 11, 12, 13, 14, 15,
                16, 17, 18, 19, 20, 21, 22, 23, 24, 25, 26, 27, 28, 29, 30, 31);
            bfrag[u] = __builtin_bit_cast(v32bf, bb);
        }
#pragma unroll
        for (int t = 0; t < 4; ++t) {
            int row = wm * 64 + t * 16 + lrow;
            const unsigned short* ap = &As[b][row * APS + lhalf * 8];
            u16x8 alo = *(const u16x8*)ap;
            u16x8 ahi = *(const u16x8*)(ap + 16);
            u16x16 aa = __builtin_shufflevector(alo, ahi,
                0, 1, 2, 3, 4, 5, 6, 7, 8, 9, 10, 11, 12, 13, 14, 15);
            v16bf afrag = __builtin_bit_cast(v16bf, aa);
            int iw = (int)AI[b][row * 2 + lhalf];
#pragma unroll
            for (int u = 0; u < 2; ++u)
                acc[t][u] = __builtin_amdgcn_swmmac_f32_16x16x64_bf16(
                    false, afrag, false, bfrag[u], acc[t][u], iw, false, false);
        }
    };

#if S24_TDM
    if (XB16) {
        // TDM staging: wave 0 moves packed-A, wave 1 moves B; DMA overlaps
        // the SWMMAC stream; padding codes reproduce APS/BPS strides.
        auto issueTdm = [&](int k0, int b) {
            if (wave == 0)
                tdm_load_2d(pw + (size_t)F0 * (K >> 1) + (k0 >> 1),
                            (unsigned)(size_t)&As[b][0],
                            32u, (unsigned)TF, (unsigned)(K >> 1), 1u, 3u, 3u);
            else if (wave == 1)
                tdm_load_2d(xhb + k0, (unsigned)(size_t)&Bs[b][0],
                            64u, (unsigned)TT, (unsigned)K, 1u, 4u, 3u);
        };
        issueTdm(0, 0); loadI(0); storeI(0);
        __builtin_amdgcn_s_wait_tensorcnt(0);
        __syncthreads();
        for (int ks = 0; ks < nk; ++ks) {
            const int cur = ks & 1;
            if (ks + 1 < nk) { issueTdm((ks + 1) * KS, cur ^ 1); loadI((ks + 1) * KS); }
            compute(cur);
            if (ks + 1 < nk) storeI(cur ^ 1);
            __builtin_amdgcn_s_wait_tensorcnt(0);
            __syncthreads();
        }
    } else
#endif
    {
        loadA(0); loadB(0); loadI(0);
        storeA(0); storeB(0); storeI(0);
        __syncthreads();
        for (int ks = 0; ks < nk; ++ks) {
            const int cur = ks & 1;
            if (ks + 1 < nk) {
                loadA((ks + 1) * KS); loadB((ks + 1) * KS); loadI((ks + 1) * KS);
            }
            compute(cur);
            if (ks + 1 < nk) { storeA(cur ^ 1); storeB(cur ^ 1); storeI(cur ^ 1); }
            __syncthreads();
        }
    }

    // epilogue: D[t][u] holds outf = F0+wm*64+t*16 + r + 8*lhalf, token = lrow
    float* po0 = out + (size_t)(T0 + wn * 32 + lrow) * N;
    float* po1 = out + (size_t)(T0 + wn * 32 + 16 + lrow) * N;
#pragma unroll
    for (int t = 0; t < 4; ++t) {
        int fb = F0 + wm * 64 + t * 16 + lhalf * 8;
#pragma unroll
        for (int r = 0; r < 8; ++r) {
            float bv = bias[fb + r];
            po0[fb + r] = scale * acc[t][0][r] + bv;
            po1[fb + r] = scale * acc[t][1][r] + bv;
        }
    }
}

// ---------------- launch ----------------

extern "C" void kernel_launch(void* const* d_in, const int* in_sizes, int n_in,
                              void* d_out, int out_size, void* d_ws, size_t ws_size,
                              hipStream_t stream) {
    const float* x    = (const float*)d_in[0];
    const float* wgt  = (const float*)d_in[1];
    const float* bias = (const float*)d_in[2];
    float* out = (float*)d_out;

    const int out_f = in_sizes[2];               // 4096
    const int in_f  = in_sizes[1] / out_f;       // 4096
    const int toks  = in_sizes[0] / in_f;        // 8192

    const int npart = out_f / 2;                 // pack blocks (2048)

    // scratch layout
    float* wsf        = (float*)d_ws;
    float* scale_ptr  = wsf;
    float* pd         = wsf + 16;
    float* ps         = pd + npart;
    size_t hdrBytes   = ((16 + 2 * (size_t)npart) * 4 + 255) & ~(size_t)255;
    unsigned short* pw = (unsigned short*)((char*)d_ws + hdrBytes);
    size_t pwHalves    = (size_t)out_f * (in_f / 2);
    unsigned* pidx     = (unsigned*)(pw + pwHalves);
    size_t idxWords    = (size_t)out_f * (in_f / 32);
    unsigned short* x16 = (unsigned short*)(pidx + idxWords);
    size_t needBase    = hdrBytes + pwHalves * 2 + idxWords * 4;
    size_t needFull    = needBase + (size_t)toks * in_f * 2;
    const bool useXh   = (ws_size >= needFull);   // ws_size fixed -> deterministic

    s24_pack_kernel<<<npart, 256, 0, stream>>>(wgt, pw, pidx, pd, ps, in_f, out_f);
    s24_scale_finalize<<<1, 256, 0, stream>>>(pd, ps, npart, scale_ptr);
    if (useXh)
        s24_xcvt_kernel<<<(int)(((size_t)toks * in_f) / 1024), 256, 0, stream>>>(x, x16);

    dim3 grid(out_f / TF, toks / TT);
    if (useXh)
        s24_swmmac_gemm<true><<<grid, 256, 0, stream>>>(
            x, x16, pw, pidx, bias, scale_ptr, out, out_f, in_f);
    else
        s24_swmmac_gemm<false><<<grid, 256, 0, stream>>>(
            x, x16, pw, pidx, bias, scale_ptr, out, out_f, in_f);
}